// ModelCopt_51977694216526
// MI455X (gfx1250) — compile-verified
//
#include <hip/hip_runtime.h>

typedef float v2f __attribute__((ext_vector_type(2)));
typedef float v8f __attribute__((ext_vector_type(8)));

#define WAVES_PER_WG 4
#define SPW 16           // samples per wave
#define HH 128
#define KK 32
#define EPSF 1e-8f

__device__ __forceinline__ v8f wmma4(v2f a, v2f b, v8f c) {
  // D = A(16x4,f32) * B(4x16,f32) + C(16x16,f32)
  return __builtin_amdgcn_wmma_f32_16x16x4_f32(false, a, false, b, (short)0, c, false, false);
}

__device__ __forceinline__ v8f zero8() {
  v8f z;
#pragma unroll
  for (int i = 0; i < 8; ++i) z[i] = 0.0f;
  return z;
}

__global__ __launch_bounds__(128, 1)
void copt_kernel(const float* __restrict__ RU,      // (B,128,3)
                 const float* __restrict__ alpha,   // (B)
                 const float* __restrict__ beta,    // (B)
                 const float* __restrict__ lam,     // (3)
                 const float* __restrict__ Bm,      // (128,32)
                 const float* __restrict__ Bdot,
                 const float* __restrict__ Bddot,
                 const float* __restrict__ Bdddot,
                 const float* __restrict__ Bpinv,   // (32,128)
                 const int*   __restrict__ steps_p,
                 const float* __restrict__ lr_p,
                 float* __restrict__ out)           // (B,32,3)
{
  // Transposed basis matrices: sBT[m][k][h] = Bmat[h][k]; pitch 129 avoids bank conflicts
  __shared__ float sBT[4][KK][HH + 1];
  __shared__ float sC [WAVES_PER_WG][3][KK][16];   // per-wave C copy (B-operand source)
  __shared__ float sG [WAVES_PER_WG][6][16][17];   // per-wave gradient fragment bounce

  const int tid  = threadIdx.x;
  const int lane = tid & 31;
  const int wave = tid >> 5;
  const int half = lane >> 4;   // 0: lanes 0-15, 1: lanes 16-31
  const int ln   = lane & 15;

  // cooperative transpose-load of the 4 basis matrices into LDS
  const float* mats[4] = {Bm, Bdot, Bddot, Bdddot};
  for (int idx = tid; idx < 4 * HH * KK; idx += 128) {
    int m = idx >> 12;
    int r = idx & 4095;
    int h = r >> 5;
    int k = r & 31;
    sBT[m][k][h] = mats[m][h * KK + k];
  }
  __syncthreads();

  const int b0   = (blockIdx.x * WAVES_PER_WG + wave) * SPW;
  const float a_n  = alpha[b0 + ln];
  const float be_n = beta[b0 + ln];
  const float l1 = lam[0], l2 = lam[1], l3 = lam[2];
  const float lr = lr_p[0];
  const int nsteps = steps_p[0];

  // ---------- init: C = Bpinv(32x128) @ RU_d(128x16) ----------
  v8f Creg[2][3];
#pragma unroll
  for (int kt = 0; kt < 2; ++kt) {
#pragma unroll
    for (int d = 0; d < 3; ++d) {
      v8f acc = zero8();
#pragma unroll 4
      for (int j = 0; j < 32; ++j) {            // K-dim = H = 128, 4 per WMMA
        int kk = 4 * j + 2 * half;
        const float* ap = Bpinv + (kt * 16 + ln) * HH + kk;
        v2f a = { ap[0], ap[1] };
        const float* rp = RU + (size_t)(b0 + ln) * (HH * 3) + kk * 3 + d;
        v2f b = { rp[0], rp[3] };
        acc = wmma4(a, b, acc);
      }
      Creg[kt][d] = acc;
#pragma unroll
      for (int i = 0; i < 8; ++i)
        sC[wave][d][kt * 16 + i + 8 * half][ln] = acc[i];
    }
  }

  // forward GEMM fragment: (16h x 16n) tile of mat @ C_d
  auto fwd = [&](int m, int ht, int d) -> v8f {
    v8f acc = zero8();
#pragma unroll
    for (int q = 0; q < 8; ++q) {               // K-dim = 32, 4 per WMMA
      int kk = 4 * q + 2 * half;
      v2f a = { sBT[m][kk][ht * 16 + ln], sBT[m][kk + 1][ht * 16 + ln] };
      v2f b = { sC[wave][d][kk][ln],      sC[wave][d][kk + 1][ln] };
      acc = wmma4(a, b, acc);
    }
    return acc;
  };

  // store a D-layout fragment into the LDS bounce tile
  auto stash = [&](int slot, const v8f& g) {
#pragma unroll
    for (int i = 0; i < 8; ++i)
      sG[wave][slot][i + 8 * half][ln] = g[i];
  };

  for (int step = 0; step < nsteps; ++step) {
    v8f dC[2][3];
#pragma unroll
    for (int kt = 0; kt < 2; ++kt)
#pragma unroll
      for (int d = 0; d < 3; ++d) dC[kt][d] = zero8();

    // backward accumulate: dC += Bᵀ_tile @ sG[slot..slot+2]
    auto bwd = [&](int m, int ht, int slot) {
#pragma unroll
      for (int kt = 0; kt < 2; ++kt)
#pragma unroll
        for (int d = 0; d < 3; ++d)
#pragma unroll
          for (int j = 0; j < 4; ++j) {         // K-dim = 16 h, 4 per WMMA
            int h0 = 4 * j + 2 * half;
            v2f a = { sBT[m][kt * 16 + ln][ht * 16 + h0],
                      sBT[m][kt * 16 + ln][ht * 16 + h0 + 1] };
            v2f b = { sG[wave][slot + d][h0][ln],
                      sG[wave][slot + d][h0 + 1][ln] };
            dC[kt][d] = wmma4(a, b, dC[kt][d]);
          }
    };

    for (int ht = 0; ht < 8; ++ht) {
      // ===== position loss: gP = 2*l1*(P - RU) =====
      {
        v8f g[3];
#pragma unroll
        for (int d = 0; d < 3; ++d) g[d] = fwd(0, ht, d);
#pragma unroll
        for (int i = 0; i < 8; ++i) {
          int h = ht * 16 + i + 8 * half;
          const float* rp = RU + (size_t)(b0 + ln) * (HH * 3) + h * 3;
#pragma unroll
          for (int d = 0; d < 3; ++d)
            g[d][i] = 2.0f * l1 * (g[d][i] - rp[d]);
        }
#pragma unroll
        for (int d = 0; d < 3; ++d) stash(d, g[d]);
        bwd(0, ht, 0);
      }
      // ===== jerk loss: gJ = 2*l3*J =====
      {
        v8f g[3];
#pragma unroll
        for (int d = 0; d < 3; ++d) g[d] = fwd(3, ht, d);
#pragma unroll
        for (int d = 0; d < 3; ++d)
#pragma unroll
          for (int i = 0; i < 8; ++i) g[d][i] *= 2.0f * l3;
#pragma unroll
        for (int d = 0; d < 3; ++d) stash(d, g[d]);
        bwd(3, ht, 0);
      }
      // ===== power-law loss: couples V and A =====
      {
        v8f gv[3], ga[3];
#pragma unroll
        for (int d = 0; d < 3; ++d) gv[d] = fwd(1, ht, d);
#pragma unroll
        for (int d = 0; d < 3; ++d) ga[d] = fwd(2, ht, d);
#pragma unroll
        for (int i = 0; i < 8; ++i) {
          float vx = gv[0][i], vy = gv[1][i], vz = gv[2][i];
          float ax = ga[0][i], ay = ga[1][i], az = ga[2][i];
          float vn  = sqrtf(vx*vx + vy*vy + vz*vz);
          float vne = vn + EPSF;
          float cx = vy*az - vz*ay;
          float cy = vz*ax - vx*az;
          float cz = vx*ay - vy*ax;
          float cn = sqrtf(cx*cx + cy*cy + cz*cz);
          float D  = vne*vne*vne + EPSF;
          float kappa = cn / D;
          float ks = fminf(fmaxf(kappa, 1e-4f), 1e4f);
          float lp_raw = be_n * logf(ks);
          float lp = fminf(fmaxf(lp_raw, -10.0f), 10.0f);
          float traw = a_n * expf(lp);
          float target = fminf(fmaxf(traw, 1e-6f), 1e6f);
          float s = vn - target;
          // VJP through the clipped chain (masks zero where clipped)
          float g_t  = -2.0f * l2 * s;
          float g_lp = (traw > 1e-6f && traw < 1e6f) ? g_t * traw : 0.0f;
          float g_ks = (lp_raw > -10.0f && lp_raw < 10.0f) ? g_lp * be_n / ks : 0.0f;
          float g_k  = (kappa > 1e-4f && kappa < 1e4f) ? g_ks : 0.0f;
          float g_cn = g_k / D;
          float g_vn = 2.0f * l2 * s - g_k * cn * 3.0f * vne * vne / (D * D);
          float icn = (cn > 0.0f) ? g_cn / cn : 0.0f;
          float gcx = icn * cx, gcy = icn * cy, gcz = icn * cz;
          float ivn = g_vn / fmaxf(vn, 1e-20f);
          // grad_V = g_vn * V/|V| + A x g_c ; grad_A = g_c x V
          float gvx = ivn * vx + (ay * gcz - az * gcy);
          float gvy = ivn * vy + (az * gcx - ax * gcz);
          float gvz = ivn * vz + (ax * gcy - ay * gcx);
          float gax = gcy * vz - gcz * vy;
          float gay = gcz * vx - gcx * vz;
          float gaz = gcx * vy - gcy * vx;
          gv[0][i] = gvx; gv[1][i] = gvy; gv[2][i] = gvz;
          ga[0][i] = gax; ga[1][i] = gay; ga[2][i] = gaz;
        }
#pragma unroll
        for (int d = 0; d < 3; ++d) { stash(d, gv[d]); stash(3 + d, ga[d]); }
        bwd(1, ht, 0);
        bwd(2, ht, 3);
      }
    } // h-tiles

    // C -= lr * dC  (registers + refresh LDS copy)
#pragma unroll
    for (int kt = 0; kt < 2; ++kt)
#pragma unroll
      for (int d = 0; d < 3; ++d) {
#pragma unroll
        for (int i = 0; i < 8; ++i) {
          float c = Creg[kt][d][i] - lr * dC[kt][d][i];
          Creg[kt][d][i] = c;
          sC[wave][d][kt * 16 + i + 8 * half][ln] = c;
        }
      }
  } // steps

  // write C (batch, 32, 3)
#pragma unroll
  for (int kt = 0; kt < 2; ++kt)
#pragma unroll
    for (int d = 0; d < 3; ++d)
#pragma unroll
      for (int i = 0; i < 8; ++i) {
        int k = kt * 16 + i + 8 * half;
        out[(size_t)(b0 + ln) * (KK * 3) + k * 3 + d] = Creg[kt][d][i];
      }
}

extern "C" void kernel_launch(void* const* d_in, const int* in_sizes, int n_in,
                              void* d_out, int out_size, void* d_ws, size_t ws_size,
                              hipStream_t stream) {
  const float* RU     = (const float*)d_in[0];
  const float* alpha  = (const float*)d_in[1];
  const float* beta   = (const float*)d_in[2];
  const float* lam    = (const float*)d_in[3];
  const float* Bm     = (const float*)d_in[4];
  const float* Bdot   = (const float*)d_in[5];
  const float* Bddot  = (const float*)d_in[6];
  const float* Bdddot = (const float*)d_in[7];
  const float* Bpinv  = (const float*)d_in[8];
  const int*   steps  = (const int*)d_in[9];
  const float* lr     = (const float*)d_in[10];
  float* out = (float*)d_out;

  int batch = in_sizes[1];                       // alpha has BATCH elements
  int nblocks = batch / (WAVES_PER_WG * SPW);    // 16384 / 64 = 256
  copt_kernel<<<nblocks, 128, 0, stream>>>(RU, alpha, beta, lam, Bm, Bdot, Bddot,
                                           Bdddot, Bpinv, steps, lr, out);
}